// AngleUpdate_80728205296196
// MI455X (gfx1250) — compile-verified
//
#include <hip/hip_runtime.h>
#include <cstdint>

typedef __attribute__((ext_vector_type(2))) float v2f;
typedef __attribute__((ext_vector_type(8))) float v8f;

#define DD 64
#define ROWS_PER_WAVE 32
#define WAVES 8
#define ROWS_PER_WG (ROWS_PER_WAVE * WAVES) /* 256 */
#define LN_EPS 1e-5f

// Sum across the 16 lanes of this lane's half-group (wave32).
__device__ __forceinline__ float sum16(float v) {
    v += __shfl_xor(v, 1, 32);
    v += __shfl_xor(v, 2, 32);
    v += __shfl_xor(v, 4, 32);
    v += __shfl_xor(v, 8, 32);
    return v;
}

__device__ __forceinline__ v8f wmma_f32(v2f a, v2f b, v8f c) {
    return __builtin_amdgcn_wmma_f32_16x16x4_f32(false, a, false, b, (short)0, c,
                                                 false, false);
}

__global__ __launch_bounds__(256)
void angle_update_kernel(const float* __restrict__ atom_feas,
                         const float* __restrict__ bond_feas,
                         const float* __restrict__ angle_feas,
                         const long long* __restrict__ bond_graph,
                         const float* __restrict__ W_core,
                         const float* __restrict__ b_core,
                         const float* __restrict__ W_gate,
                         const float* __restrict__ b_gate,
                         const float* __restrict__ g1, const float* __restrict__ be1,
                         const float* __restrict__ g2, const float* __restrict__ be2,
                         const float* __restrict__ g3, const float* __restrict__ be3,
                         float* __restrict__ out, int n) {
    // Weights in LDS as K-row pairs: ldsX[p*64 + (n ^ ((p&1)<<4))] = {W[2p][n], W[2p+1][n]}
    __shared__ v2f ldsC[8192]; // 64 KB
    __shared__ v2f ldsG[8192]; // 64 KB

    const int tid = threadIdx.x;
    for (int i = tid; i < 8192; i += 256) {
        const int p = i >> 6, nn = i & 63;
        const int ns = nn ^ ((p & 1) << 4);
        v2f w;
        w.x = W_core[(2 * p) * DD + nn];
        w.y = W_core[(2 * p + 1) * DD + nn];
        ldsC[p * DD + ns] = w;
        w.x = W_gate[(2 * p) * DD + nn];
        w.y = W_gate[(2 * p + 1) * DD + nn];
        ldsG[p * DD + ns] = w;
    }
    __syncthreads();

    const int lane = tid & 31;
    const int wave = tid >> 5;
    const int half = lane >> 4; // 0: K even pair, 1: K odd pair
    const int lc = lane & 15;   // row-in-tile for A, col-in-tile for B/C/D

    const long long rowbase =
        (long long)blockIdx.x * ROWS_PER_WG + (long long)wave * ROWS_PER_WAVE;

    // Per-lane gathered row base pointers for the 4 input segments, per tile.
    const float* aptr[2][4];
#pragma unroll
    for (int t = 0; t < 2; ++t) {
        long long r = rowbase + t * 16 + lc;
        if (r > (long long)n - 1) r = (long long)n - 1;
        const long long i0 = bond_graph[r * 5 + 0];
        const long long i1 = bond_graph[r * 5 + 1];
        const long long i2 = bond_graph[r * 5 + 2];
        aptr[t][0] = bond_feas + i1 * DD;   // bond_i
        aptr[t][1] = bond_feas + i2 * DD;   // bond_j
        aptr[t][2] = angle_feas + r * DD;   // angle
        aptr[t][3] = atom_feas + i0 * DD;   // center atom
    }

    v8f accC[2][4], accG[2][4];
    const v8f vzero = {0.f, 0.f, 0.f, 0.f, 0.f, 0.f, 0.f, 0.f};
#pragma unroll
    for (int t = 0; t < 2; ++t)
#pragma unroll
        for (int c = 0; c < 4; ++c) { accC[t][c] = vzero; accG[t][c] = vzero; }

    // Main GEMM loop: K = 256 = 4 segments x 16 steps of K=4.
#pragma unroll
    for (int seg = 0; seg < 4; ++seg) {
        const float* p0 = aptr[0][seg] + 2 * half;
        const float* p1 = aptr[1][seg] + 2 * half;
#pragma unroll 4
        for (int s = 0; s < 16; ++s) {
            const v2f a0 = *(const v2f*)(p0 + 4 * s);
            const v2f a1 = *(const v2f*)(p1 + 4 * s);
            const int pr = seg * 32 + 2 * s + half; // K-pair index (pr & 1 == half)
            const int base = pr * DD;
#pragma unroll
            for (int c = 0; c < 4; ++c) {
                const int ns = (16 * c + lc) ^ (half << 4);
                const v2f bc = ldsC[base + ns];
                accC[0][c] = wmma_f32(a0, bc, accC[0][c]);
                accC[1][c] = wmma_f32(a1, bc, accC[1][c]);
                const v2f bg = ldsG[base + ns];
                accG[0][c] = wmma_f32(a0, bg, accG[0][c]);
                accG[1][c] = wmma_f32(a1, bg, accG[1][c]);
            }
        }
    }

    // Per-lane column parameters (column = 16*c + lc).
    float bcv[4], bgv[4], g1v[4], be1v[4], g2v[4], be2v[4], g3v[4], be3v[4];
#pragma unroll
    for (int c = 0; c < 4; ++c) {
        const int col = 16 * c + lc;
        bcv[c] = b_core[col]; bgv[c] = b_gate[col];
        g1v[c] = g1[col];     be1v[c] = be1[col];
        g2v[c] = g2[col];     be2v[c] = be2[col];
        g3v[c] = g3[col];     be3v[c] = be3[col];
    }

    // Epilogue: LN1 -> silu, LN2 -> sigmoid, residual, LN3, store.
    // C/D layout: reg r holds row M = r + 8*half, column N = 16*c + lc.
#pragma unroll
    for (int t = 0; t < 2; ++t) {
        const long long mrow = rowbase + t * 16 + half * 8;
#pragma unroll
        for (int r = 0; r < 8; ++r) {
            float xc[4], xg[4];
            float s1 = 0.f, q1 = 0.f, s2 = 0.f, q2 = 0.f;
#pragma unroll
            for (int c = 0; c < 4; ++c) {
                const float v = accC[t][c][r] + bcv[c];
                xc[c] = v; s1 += v; q1 += v * v;
                const float w = accG[t][c][r] + bgv[c];
                xg[c] = w; s2 += w; q2 += w * w;
            }
            s1 = sum16(s1); q1 = sum16(q1);
            s2 = sum16(s2); q2 = sum16(q2);
            const float mu1 = s1 * (1.f / 64.f);
            const float rs1 = rsqrtf(q1 * (1.f / 64.f) - mu1 * mu1 + LN_EPS);
            const float mu2 = s2 * (1.f / 64.f);
            const float rs2 = rsqrtf(q2 * (1.f / 64.f) - mu2 * mu2 + LN_EPS);

            const long long grow = mrow + r;
            const bool valid = grow < (long long)n;
            const long long gr = valid ? grow : (long long)n - 1;

            float nv[4];
            float s3 = 0.f, q3 = 0.f;
#pragma unroll
            for (int c = 0; c < 4; ++c) {
                const float y1 = (xc[c] - mu1) * rs1 * g1v[c] + be1v[c];
                const float core = y1 / (1.f + __expf(-y1)); // silu
                const float y2 = (xg[c] - mu2) * rs2 * g2v[c] + be2v[c];
                const float gate = 1.f / (1.f + __expf(-y2)); // sigmoid
                const float ang = angle_feas[gr * DD + 16 * c + lc];
                const float v = core * gate + ang;
                nv[c] = v; s3 += v; q3 += v * v;
            }
            s3 = sum16(s3); q3 = sum16(q3);
            const float mu3 = s3 * (1.f / 64.f);
            const float rs3 = rsqrtf(q3 * (1.f / 64.f) - mu3 * mu3 + LN_EPS);

            if (valid) {
#pragma unroll
                for (int c = 0; c < 4; ++c)
                    out[grow * DD + 16 * c + lc] =
                        (nv[c] - mu3) * rs3 * g3v[c] + be3v[c];
            }
        }
    }
}

extern "C" void kernel_launch(void* const* d_in, const int* in_sizes, int n_in,
                              void* d_out, int out_size, void* d_ws, size_t ws_size,
                              hipStream_t stream) {
    const float* atom_feas = (const float*)d_in[0];
    const float* bond_feas = (const float*)d_in[1];
    const float* angle_feas = (const float*)d_in[2];
    const long long* bond_graph = (const long long*)d_in[3]; // int64
    const float* W_core = (const float*)d_in[4];
    const float* b_core = (const float*)d_in[5];
    const float* W_gate = (const float*)d_in[6];
    const float* b_gate = (const float*)d_in[7];
    const float* g1 = (const float*)d_in[8];
    const float* be1 = (const float*)d_in[9];
    const float* g2 = (const float*)d_in[10];
    const float* be2 = (const float*)d_in[11];
    const float* g3 = (const float*)d_in[12];
    const float* be3 = (const float*)d_in[13];

    const int n = in_sizes[3] / 5; // n_angles
    const int blocks = (n + ROWS_PER_WG - 1) / ROWS_PER_WG;

    angle_update_kernel<<<blocks, 256, 0, stream>>>(
        atom_feas, bond_feas, angle_feas, bond_graph, W_core, b_core, W_gate,
        b_gate, g1, be1, g2, be2, g3, be3, (float*)d_out, n);
}